// relational_model_70248485093750
// MI455X (gfx1250) — compile-verified
//
#include <hip/hip_runtime.h>

// Problem constants (from reference setup_inputs)
#define S_LEN  2048
#define D_DIM  768
#define M_COMP 128
#define R_REL  4
#define RD     (R_REL * D_DIM)   // 3072
#define BATCH  16

// LDS padded strides (64 banks x 4B): keep fragment gathers conflict-free
#define FJ_STRIDE 772            // 768 + 4
#define TJ_STRIDE 3076           // 3072 + 4
#define NCHUNK    128            // W columns staged per chunk (8 waves x 16)
#define KPANEL    64             // W rows per async panel
#define WP_STRIDE 132            // 128 + 4  (TDM pad: 128 dwords + 4 dwords)
#define N_PANELS  (D_DIM / KPANEL)   // 12

typedef float        v2f __attribute__((ext_vector_type(2)));
typedef float        v8f __attribute__((ext_vector_type(8)));
typedef int          v4i __attribute__((ext_vector_type(4)));
typedef int          v8i __attribute__((ext_vector_type(8)));
typedef unsigned int v4u __attribute__((ext_vector_type(4)));

typedef __attribute__((address_space(1))) v4i g_v4i;   // global int4
typedef __attribute__((address_space(3))) v4i l_v4i;   // LDS int4

#if __has_builtin(__builtin_amdgcn_global_load_async_to_lds_b128)
#define HAS_ASYNC_LDS 1
#else
#define HAS_ASYNC_LDS 0
#endif

#if __has_builtin(__builtin_amdgcn_tensor_load_to_lds) && \
    __has_builtin(__builtin_amdgcn_s_wait_tensorcnt)
#define HAS_TDM 1
#else
#define HAS_TDM 0
#endif

// 16-byte global -> LDS copy (async on CDNA5, sync fallback otherwise)
__device__ __forceinline__ void copy16_g2l(const float* g, float* l) {
#if HAS_ASYNC_LDS
  const v4i* gg = reinterpret_cast<const v4i*>(g);
  v4i*       ll = reinterpret_cast<v4i*>(l);
  __builtin_amdgcn_global_load_async_to_lds_b128(
      (g_v4i*)gg, (l_v4i*)ll, 0, 0);
#else
  *reinterpret_cast<float4*>(l) = *reinterpret_cast<const float4*>(g);
#endif
}

__device__ __forceinline__ void async_wait_le8() {
#if HAS_ASYNC_LDS
  asm volatile("s_wait_asynccnt 0x8" ::: "memory");
#endif
}
__device__ __forceinline__ void async_wait_0() {
#if HAS_ASYNC_LDS
  asm volatile("s_wait_asynccnt 0x0" ::: "memory");
#endif
}

// LDS byte offset of a generic shared-memory pointer (AS3 ptrtoint)
__device__ __forceinline__ unsigned lds_off_of(const void* p) {
  return (unsigned)(size_t)(const __attribute__((address_space(3))) char*)p;
}

#if HAS_TDM
// TDM: move one 64-row x 512B panel of W into LDS with 16B padding per row
// (hardware-applied stride 132 dwords). D# per CDNA5 ISA ch. 8.
// This toolchain exposes the 6-arg builtin:
//   (uint32x4 g0, int32x8 g1, int32x4 g2, int32x4 g3, int32x8 pad, i32 cpol)
__device__ __forceinline__ void tdm_load_panel(const float* gsrc,
                                               unsigned lds_byte_off) {
  const unsigned long long ga = (unsigned long long)(size_t)gsrc;
  v4u g0;
  g0[0] = 1u;                                            // count=1 (user D#)
  g0[1] = lds_byte_off;                                  // lds_addr
  g0[2] = (unsigned)(ga & 0xFFFFFFFFu);                  // global_addr lo
  g0[3] = (unsigned)((ga >> 32) & 0x01FFFFFFu)           // global_addr hi
        | (2u << 30);                                    // type = 2 (image)
  v8i g1;
  g1[0] = (2 << 16)                                      // data_size = 4B
        | (1 << 20)                                      // pad_enable
        | (6 << 22)                                      // pad_interval: 128 dw
        | (3 << 25);                                     // pad_amount: 4 dw
  g1[1] = 0;                                             // tensor_dim0 lo16 = 0
  g1[2] = 0x4000;                                        // tensor_dim0 = 1<<30
  g1[3] = 0x4000 | (128 << 16);                          // dim1 = 1<<30, tile_dim0 = 128
  g1[4] = KPANEL;                                        // tile_dim1 = 64, tile_dim2 = 0
  g1[5] = RD;                                            // tensor_dim0_stride = 3072
  g1[6] = 0;
  g1[7] = 0;
  v4i z4 = {0, 0, 0, 0};                                 // groups 2/3: <=2D tensor
  v8i z8 = {0, 0, 0, 0, 0, 0, 0, 0};
  __builtin_amdgcn_tensor_load_to_lds(g0, g1, z4, z4, z8, 0);
}
#endif

// ---------------------------------------------------------------------------
// Kernel A: segment means -> F [16][128][768], valid mask [16][128]
// grid (16 batches, 6 column-chunks of 128), block 256
// ---------------------------------------------------------------------------
__global__ void __launch_bounds__(256)
seg_mean_kernel(const float* __restrict__ embds,
                const int* __restrict__ cmask,
                float* __restrict__ F,
                int* __restrict__ valid) {
  const int b = blockIdx.x;
  const int colBase = blockIdx.y * 128;
  const int t = threadIdx.x;

  __shared__ int s_scan[256];
  __shared__ int s_pos[128];   // first 127 chosen positions

  // ---- phase 1: positions of first 127 chosen components -----------------
  const int* cm = cmask + b * S_LEN;
  int mloc[8];
  int cnt = 0;
#pragma unroll
  for (int k = 0; k < 8; ++k) {
    mloc[k] = (cm[t * 8 + k] != 0);
    cnt += mloc[k];
  }
  s_scan[t] = cnt;
  __syncthreads();
  for (int d = 1; d < 256; d <<= 1) {          // inclusive scan
    int v = (t >= d) ? s_scan[t - d] : 0;
    __syncthreads();
    s_scan[t] += v;
    __syncthreads();
  }
  int rank = s_scan[t] - cnt;                  // exclusive prefix
#pragma unroll
  for (int k = 0; k < 8; ++k) {
    if (mloc[k]) {
      if (rank < 127) s_pos[rank] = t * 8 + k;
      rank++;
    }
  }
  __syncthreads();
  const int numChosen = s_scan[255];

  // ---- phase 2: write F --------------------------------------------------
  float* Fb = F + (size_t)b * (M_COMP * D_DIM);

  if (t < 32) {                                 // root row = 1/D
    const float rv = 1.0f / (float)D_DIM;
    float4 rootv = make_float4(rv, rv, rv, rv);
    *reinterpret_cast<float4*>(Fb + colBase + t * 4) = rootv;
  }
  if (blockIdx.y == 0 && t < M_COMP) {
    valid[b * M_COMP + t] = ((t == 0) || ((t - 1) < numChosen)) ? 1 : 0;
  }

  const int wave = t >> 5;
  const int lane = t & 31;
  const float* Eb = embds + (size_t)b * S_LEN * D_DIM;

  for (int j = wave; j < 127; j += 8) {
    const int up  = (j < numChosen) ? (s_pos[j] + 1) : 0;
    const int low = (j == 0) ? 0 : (((j - 1) < numChosen) ? (s_pos[j - 1] + 1) : 0);
    float4 acc = make_float4(0.f, 0.f, 0.f, 0.f);
    for (int row = low; row < up; ++row) {
      __builtin_prefetch(Eb + (size_t)(row + 8) * D_DIM + colBase + lane * 4, 0, 0);
      const float4 v = *reinterpret_cast<const float4*>(
          Eb + (size_t)row * D_DIM + colBase + lane * 4);
      acc.x += v.x; acc.y += v.y; acc.z += v.z; acc.w += v.w;
    }
    const float sc = (up > low) ? (1.0f / (float)(up - low)) : 0.0f;
    float4 o = make_float4(acc.x * sc, acc.y * sc, acc.z * sc, acc.w * sc);
    *reinterpret_cast<float4*>(Fb + (size_t)(j + 1) * D_DIM + colBase + lane * 4) = o;
  }
}

// ---------------------------------------------------------------------------
// Kernel B: fused  T_j = F_j @ W  then  E = F @ T_j^T  (+ masking)
// grid (16 batches, 8 j-tiles of 16), block 256 (8 waves)
// LDS: Fj[16][772] + Tj[16][3076] + Wpanel[2][64][132]  = 313856 B
// W panels staged by the Tensor Data Mover (double buffered), fallback async.
// ---------------------------------------------------------------------------
__global__ void __launch_bounds__(256)
energy_kernel(const float* __restrict__ F,
              const float* __restrict__ W,
              const int* __restrict__ valid,
              float* __restrict__ out) {
  const int b  = blockIdx.x;
  const int jt = blockIdx.y;      // 16 j-rows per tile
  const int t  = threadIdx.x;
  const int wave = t >> 5;
  const int lane = t & 31;
  const int r  = lane & 15;       // lane within half-wave
  const int hi = lane >> 4;       // 0 or 1 (K/M half selector per WMMA layout)

  extern __shared__ float smem[];
  float* Fj = smem;                                   // [16][FJ_STRIDE]
  float* Tj = smem + 16 * FJ_STRIDE;                  // [16][TJ_STRIDE]
  float* Wp = smem + 16 * FJ_STRIDE + 16 * TJ_STRIDE; // [2][KPANEL][WP_STRIDE]

  const float* Fb  = F + (size_t)b * (M_COMP * D_DIM);
  const float* Fjg = Fb + (size_t)jt * 16 * D_DIM;

  // stage F_j into LDS (async, padded row stride)
  for (int i = t; i < 16 * (D_DIM / 16); i += 256) {  // 768 x 16B chunks
    const int row = i / 48;                            // 48 chunks per row
    const int c4  = (i % 48) * 4;                      // float offset
    copy16_g2l(Fjg + (size_t)row * D_DIM + c4, Fj + row * FJ_STRIDE + c4);
  }
  async_wait_0();
  __syncthreads();

  // ---- step 1: Tj[16,3072] = Fj[16,768] @ W[768,3072] ---------------------
  const int myN = wave * 16;                     // wave's 16 cols in the chunk
  for (int nc = 0; nc < RD / NCHUNK; ++nc) {
    const int ncBase = nc * NCHUNK;

#if HAS_TDM
    auto issue_panel = [&](int p, int buf) {     // one TDM op per panel
      if (wave == 0) {
        tdm_load_panel(W + (size_t)(p * KPANEL) * RD + ncBase,
                       lds_off_of(Wp + buf * (KPANEL * WP_STRIDE)));
      }
    };
#else
    auto issue_panel = [&](int p, int buf) {     // per-lane async fallback
      const int kbase = p * KPANEL;
      float* dst = Wp + buf * (KPANEL * WP_STRIDE);
#pragma unroll
      for (int i = 0; i < 8; ++i) {              // 2048 x 16B chunks total
        const int f  = t + i * 256;
        const int k  = f >> 5;                   // 0..63
        const int c4 = (f & 31) * 4;             // 0..124 floats
        copy16_g2l(W + (size_t)(kbase + k) * RD + ncBase + c4,
                   dst + k * WP_STRIDE + c4);
      }
    };
#endif

    issue_panel(0, 0);
    v8f acc = {0.f, 0.f, 0.f, 0.f, 0.f, 0.f, 0.f, 0.f};

    for (int p = 0; p < N_PANELS; ++p) {
      if (p + 1 < N_PANELS) {
        issue_panel(p + 1, (p + 1) & 1);
#if HAS_TDM
        if (wave == 0) __builtin_amdgcn_s_wait_tensorcnt(1);
#else
        async_wait_le8();
#endif
      } else {
#if HAS_TDM
        if (wave == 0) __builtin_amdgcn_s_wait_tensorcnt(0);
#else
        async_wait_0();
#endif
      }
      __syncthreads();                           // panel p visible to all

      const float* Wpan = Wp + (p & 1) * (KPANEL * WP_STRIDE);
      const int kbase = p * KPANEL;
#pragma unroll 4
      for (int kl = 0; kl < KPANEL; kl += 4) {
        const int kk = kl + 2 * hi;
        v2f a, bw;
        a.x  = Fj[r * FJ_STRIDE + kbase + kk];
        a.y  = Fj[r * FJ_STRIDE + kbase + kk + 1];
        bw.x = Wpan[kk * WP_STRIDE + myN + r];
        bw.y = Wpan[(kk + 1) * WP_STRIDE + myN + r];
        acc = __builtin_amdgcn_wmma_f32_16x16x4_f32(
            false, a, false, bw, (short)0, acc, false, false);
      }
      __syncthreads();                           // buffer reuse fence
    }

#pragma unroll
    for (int v = 0; v < 8; ++v) {
      Tj[(v + 8 * hi) * TJ_STRIDE + ncBase + myN + r] = acc[v];
    }
  }
  __syncthreads();

  // ---- step 2: E[i, (jl,rr)] = sum_d F[i,d] * Tj[jl, d*4+rr]  + masking ---
  for (int q = 0; q < 4; ++q) {
    const int tile = wave + 8 * q;        // 32 tiles of 16x16 over [128 x 64]
    const int i0 = (tile >> 2) * 16;
    const int c0 = (tile & 3) * 16;
    const int c  = c0 + r;                // column = jl*4 + rr
    const int jl = c >> 2;
    const int rr = c & 3;

    v8f acc = {0.f, 0.f, 0.f, 0.f, 0.f, 0.f, 0.f, 0.f};
#pragma unroll 4
    for (int k0 = 0; k0 < D_DIM; k0 += 4) {
      const int kk = k0 + 2 * hi;
      v2f a, bb;
      a.x  = Fb[(size_t)(i0 + r) * D_DIM + kk];
      a.y  = Fb[(size_t)(i0 + r) * D_DIM + kk + 1];
      bb.x = Tj[jl * TJ_STRIDE + kk * R_REL + rr];
      bb.y = Tj[jl * TJ_STRIDE + (kk + 1) * R_REL + rr];
      acc = __builtin_amdgcn_wmma_f32_16x16x4_f32(
          false, a, false, bb, (short)0, acc, false, false);
    }

    const int  jg = jt * 16 + jl;
    const bool vj = (valid[b * M_COMP + jg] != 0);
#pragma unroll
    for (int v = 0; v < 8; ++v) {
      const int  i  = i0 + v + 8 * hi;
      const bool ok = vj && (i != 0) && (i != jg) && (valid[b * M_COMP + i] != 0);
      const float val = ok ? acc[v] : -__builtin_inff();
      out[(((size_t)b * M_COMP + i) * M_COMP + jg) * R_REL + rr] = val;
    }
  }
}

// ---------------------------------------------------------------------------
// Host launcher
// ---------------------------------------------------------------------------
extern "C" void kernel_launch(void* const* d_in, const int* in_sizes, int n_in,
                              void* d_out, int out_size, void* d_ws, size_t ws_size,
                              hipStream_t stream) {
  const float* embds = (const float*)d_in[0];   // [16,2048,768] f32
  const int*   cmask = (const int*)d_in[1];     // [16,2048] i32
  const float* W     = (const float*)d_in[2];   // [768,3072] f32
  // d_in[3]=max_comps(128), d_in[4]=n_rels(4) are compile-time constants here.

  float* Fws    = (float*)d_ws;                                    // 6.29 MB
  int*   validw = (int*)((char*)d_ws +
                  (size_t)BATCH * M_COMP * D_DIM * sizeof(float)); // 8 KB
  float* out = (float*)d_out;

  seg_mean_kernel<<<dim3(BATCH, D_DIM / 128), 256, 0, stream>>>(
      embds, cmask, Fws, validw);

  const size_t smemB =
      (size_t)(16 * FJ_STRIDE + 16 * TJ_STRIDE + 2 * KPANEL * WP_STRIDE) *
      sizeof(float);                                               // 313856 B
  (void)hipFuncSetAttribute((const void*)energy_kernel,
                            hipFuncAttributeMaxDynamicSharedMemorySize,
                            (int)smemB);
  energy_kernel<<<dim3(BATCH, M_COMP / 16), 256, smemB, stream>>>(
      Fws, W, validw, out);
}